// Conv2d_27281632264865
// MI455X (gfx1250) — compile-verified
//
#include <hip/hip_runtime.h>

// CDNA5 (gfx1250) wave32 WMMA conv3x3 as implicit GEMM.
// LDS im2col layout interleaves K-pairs so every WMMA A/B operand is a single
// conflict-free ds_load_b64 into an adjacent VGPR pair. V_WMMA_F32_16X16X4_F32.

typedef __attribute__((ext_vector_type(2))) float v2f;
typedef __attribute__((ext_vector_type(8))) float v8f;

#define CIN   64
#define COUT  64
#define HH    112
#define WW    112
#define KTOT  576              // CIN * 9
#define KC    48               // K elements per staged chunk (multiple of 4)
#define NPAIR (KC / 2)         // 24 K-pairs per chunk
#define BSTR  288              // floats per K-pair row of ldsB: 128 px * 2 + 32 pad (288%64==32)
#define WSTR  160              // floats per K-pair row of ldsW:  64 co * 2 + 32 pad (160%64==32)

__global__ __launch_bounds__(256) void conv3x3_wmma_f32(
    const float* __restrict__ image,    // [32][64][112][112]
    const float* __restrict__ weight,   // [64][64][3][3] == [o][k], k = ci*9+ky*3+kx
    const float* __restrict__ bias,     // [64]
    float* __restrict__ out)            // [32][64][112][112]
{
    __shared__ float ldsB[NPAIR * BSTR];  // 27.6 KB: im2col chunk [pair][pixel][2]
    __shared__ float ldsW[NPAIR * WSTR];  // 15.4 KB: weights chunk [pair][cout][2]

    const int tid  = threadIdx.x;
    const int lane = tid & 31;
    const int wave = tid >> 5;
    const int n16  = lane & 15;   // B col = x position; A row = cout
    const int hi   = lane >> 4;   // selects K pair {0,1} vs {2,3} (ISA 16x4 layout)

    const int xg    = blockIdx.x;            // 0..6   (112 = 7*16)
    const int yg    = blockIdx.y;            // 0..13  (112 = 14*8)
    const int img_n = blockIdx.z;            // 0..31

    const int coutBase = (wave & 3) * 16;    // 4 cout tiles
    const int pg       = wave >> 2;          // 2 row groups of 4 rows
    const int x0       = xg * 16;
    const int y0       = yg * 8;
    const int yrow0    = y0 + pg * 4;

    const float* imgN = image + (size_t)img_n * (CIN * HH * WW);

    // ---- thread-fixed im2col staging coordinates ----
    const int pixS  = tid & 127;        // pixel slot: row*16 + x
    const int rowS  = pixS >> 4;        // 0..7
    const int xS    = pixS & 15;
    const int eB    = tid >> 7;         // kl parity (0 or 1)
    const int rowBase = y0 - 1 + rowS;
    const int xBase   = x0 - 1 + xS;
    // base-3 digits (ci,ky,kx) of this thread's staging k = eB; advance by +2 each step
    int bci = 0, bky = 0, bkx = eB;

    v8f acc[4] = {};

    for (int chunk = 0; chunk < KTOT / KC; ++chunk) {
        const int kbase = chunk * KC;

        __syncthreads();   // previous chunk fully consumed before overwrite

        // ---- stage weights chunk: ldsW[kl>>1][o][kl&1] ----
        for (int i = tid; i < COUT * KC; i += 256) {
            const int o  = i & 63;
            const int kl = i >> 6;
            ldsW[(kl >> 1) * WSTR + o * 2 + (kl & 1)] = weight[o * KTOT + kbase + kl];
        }

        // ---- stage im2col chunk: ldsB[j][pix][eB], kl = eB + 2*j ----
        for (int j = 0; j < NPAIR; ++j) {
            const int gy = rowBase + bky;
            const int gx = xBase + bkx;
            float v = 0.0f;
            if ((unsigned)gy < HH && (unsigned)gx < WW)
                v = imgN[bci * (HH * WW) + gy * WW + gx];
            ldsB[j * BSTR + pixS * 2 + eB] = v;
            // advance k by 2 (base-3 carry over kx,ky -> ci)
            bkx += 2;
            if (bkx > 2) { bkx -= 3; bky += 1; if (bky > 2) { bky -= 3; bci += 1; } }
        }

        __syncthreads();

        // Prefetch next weights chunk toward L2 (gfx1250 global_prefetch_b8).
        if (chunk + 1 < KTOT / KC) {
            __builtin_prefetch(&weight[(tid & 63) * KTOT + kbase + KC], 0, 1);
        }

        // ---- inner GEMM loop: 2 adds + 5 ds_load_b64 + 4 WMMA per K-step ----
        for (int step = 0; step < KC / 4; ++step) {
            const int p = 2 * step + hi;   // K-pair index for this lane half

            const v2f a = *(const v2f*)(ldsW + p * WSTR + n16 * 2);
            const float* bp = ldsB + p * BSTR + pg * 128 + n16 * 2;

            #pragma unroll
            for (int r = 0; r < 4; ++r) {
                const v2f b = *(const v2f*)(bp + r * 32);
                acc[r] = __builtin_amdgcn_wmma_f32_16x16x4_f32(
                    /*neg_a=*/false, a, /*neg_b=*/false, b,
                    /*c_mod=*/(short)0, acc[r],
                    /*reuse_a=*/false, /*reuse_b=*/false);
            }
        }
    }

    // ---- bias add + coalesced stores: lane%16 = x, VGPR g + 8*hi = cout ----
    float* outN = out + (size_t)img_n * (COUT * HH * WW);
    #pragma unroll
    for (int r = 0; r < 4; ++r) {
        const int y = yrow0 + r;
        #pragma unroll
        for (int g = 0; g < 8; ++g) {
            const int co = coutBase + 8 * hi + g;
            const float v = acc[r][g] + bias[co];
            outN[co * (HH * WW) + y * WW + x0 + n16] = v;
        }
    }
}

extern "C" void kernel_launch(void* const* d_in, const int* in_sizes, int n_in,
                              void* d_out, int out_size, void* d_ws, size_t ws_size,
                              hipStream_t stream) {
    const float* image  = (const float*)d_in[0];  // 32*64*112*112
    const float* weight = (const float*)d_in[1];  // 64*64*3*3
    const float* bias   = (const float*)d_in[2];  // 64
    float* out          = (float*)d_out;          // 32*64*112*112

    dim3 grid(WW / 16, HH / 8, 32);   // (7, 14, 32)
    dim3 block(256);                  // 8 wave32 waves
    conv3x3_wmma_f32<<<grid, block, 0, stream>>>(image, weight, bias, out);
}